// MoESSMBlock_50362786512976
// MI455X (gfx1250) — compile-verified
//
#include <hip/hip_runtime.h>
#include <math.h>

// Problem dims (fixed by the reference)
#define EMBED  512
#define NEXP   5
#define DSTATE 64
#define DCONV  4
#define DIN    1024
#define DTRANK 32
#define BB     2
#define LL     256
#define NTOK   (BB * LL)      // 512 tokens
#define KC     32             // K-chunk staged in LDS per GEMM iteration

typedef __attribute__((ext_vector_type(2))) float v2f;
typedef __attribute__((ext_vector_type(4))) float v4f;
typedef __attribute__((ext_vector_type(8))) float v8f;
typedef int v4i __attribute__((vector_size(16)));     // matches builtin's param type

#define AS_GLOBAL __attribute__((address_space(1)))
#define AS_LOCAL  __attribute__((address_space(3)))

__device__ __forceinline__ float silu_f(float x)     { return x / (1.f + __expf(-x)); }
__device__ __forceinline__ float softplus_f(float x) { return (x > 20.f) ? x : log1pf(__expf(x)); }
__device__ __forceinline__ float gelu_f(float x)     { return 0.5f * x * (1.f + erff(x * 0.70710678118654752f)); }

// ---- CDNA5 async global->LDS staging (ASYNCcnt path) ----------------------
__device__ __forceinline__ void async_copy_b128(const float* g, float* l) {
#if __has_builtin(__builtin_amdgcn_global_load_async_to_lds_b128)
  __builtin_amdgcn_global_load_async_to_lds_b128(
      (AS_GLOBAL v4i*)(v4i*)(void*)const_cast<float*>(g),
      (AS_LOCAL  v4i*)(v4i*)(void*)l,
      /*offset=*/0, /*cpol=*/0);
#else
  *(v4f*)l = *(const v4f*)g;   // fallback: synchronous copy through VGPRs
#endif
}

__device__ __forceinline__ void wait_async0() {
#if __has_builtin(__builtin_amdgcn_s_wait_asynccnt)
  __builtin_amdgcn_s_wait_asynccnt(0);
#else
  asm volatile("s_wait_asynccnt 0x0" ::: "memory");
#endif
}

__device__ __forceinline__ v8f wmma_f32(v2f a, v2f b, v8f c) {
  return __builtin_amdgcn_wmma_f32_16x16x4_f32(false, a, false, b, (short)0, c, false, false);
}

__device__ __forceinline__ void store_c(float* __restrict__ C, const float* __restrict__ addend,
                                        const float* __restrict__ bias, int ldc,
                                        int row, int col, int N, int act, float v) {
  if (col >= N) return;
  if (bias) v += bias[col];
  if (act == 1)      v = softplus_f(v);
  else if (act == 2) v = gelu_f(v);
  if (addend) v += addend[(size_t)row * ldc + col];
  C[(size_t)row * ldc + col] = v;
}

// ---------------------------------------------------------------------------
// WMMA-f32 GEMM: C[M,N] = act(A[M,K] * W[N,K]^T + bias[N]) + addend
//  - Block: 128 threads = 2x2 waves; block tile 64x64, wave tile 32x32
//    (2 A-frags x 2 B-frags -> 4 x V_WMMA_F32_16X16X4_F32 per K-step).
//  - A(64xKC) and B(64xKC) chunks double-buffered in LDS, filled with
//    global_load_async_to_lds_b128 (ASYNCcnt), prefetching chunk i+1 while
//    computing chunk i from the other buffer.
//  - M, K must be multiples of 64/KC (true here); N handled with predication.
//  act: 0=none, 1=softplus, 2=gelu
// ---------------------------------------------------------------------------
__global__ void wmma_gemm_kernel(const float* __restrict__ A, int lda,
                                 const float* __restrict__ W, int ldw,
                                 const float* __restrict__ bias,
                                 const float* __restrict__ addend,
                                 float* __restrict__ C, int ldc,
                                 int N, int K, int act) {
  __shared__ float As[2][64 * KC];
  __shared__ float Bs[2][64 * KC];

  const int tid  = threadIdx.x;
  const int lane = tid & 31;
  const int wave = tid >> 5;
  const int wm   = wave >> 1, wn = wave & 1;
  const int half = lane >> 4, l15 = lane & 15;

  const int mbase = blockIdx.y * 64;
  const int nbase = blockIdx.x * 64;

  // One staging pass: 128 threads x 4 iters x b128 covers a 64xKC chunk.
  auto stage = [&](int kc, int buf) {
#pragma unroll
    for (int t = 0; t < 4; ++t) {
      int q   = tid + t * 128;           // 0..511
      int row = q >> 3;                  // 0..63
      int c4  = (q & 7) * 4;             // 0,4,...,28
      async_copy_b128(A + (size_t)(mbase + row) * lda + kc + c4,
                      &As[buf][row * KC + c4]);
      int n = nbase + row; if (n > N - 1) n = N - 1;   // clamp (stores predicated)
      async_copy_b128(W + (size_t)n * ldw + kc + c4,
                      &Bs[buf][row * KC + c4]);
    }
  };

  v8f acc00 = {}, acc01 = {}, acc10 = {}, acc11 = {};
  const int m0l = wm * 32 + l15, m1l = m0l + 16;   // local A rows for the 2 M-frags
  const int n0l = wn * 32 + l15, n1l = n0l + 16;   // local B rows for the 2 N-frags

  stage(0, 0);
  const int nch = K / KC;
  for (int i = 0; i < nch; ++i) {
    wait_async0();        // chunk i landed (own wave's async ops)
    __syncthreads();      // ...and everyone else's
    if (i + 1 < nch) stage((i + 1) * KC, (i + 1) & 1);
    const float* Ab = As[i & 1];
    const float* Bb = Bs[i & 1];
#pragma unroll
    for (int k = 0; k < KC; k += 4) {
      v2f a0 = *(const v2f*)&Ab[m0l * KC + k + half * 2];
      v2f a1 = *(const v2f*)&Ab[m1l * KC + k + half * 2];
      v2f b0 = *(const v2f*)&Bb[n0l * KC + k + half * 2];
      v2f b1 = *(const v2f*)&Bb[n1l * KC + k + half * 2];
      acc00 = wmma_f32(a0, b0, acc00);
      acc01 = wmma_f32(a0, b1, acc01);
      acc10 = wmma_f32(a1, b0, acc10);
      acc11 = wmma_f32(a1, b1, acc11);
    }
    __syncthreads();      // all waves done with buf[i&1] before it is re-staged
  }

  const int r0 = mbase + wm * 32 + half * 8;
  const int c0 = nbase + wn * 32 + l15;
#pragma unroll
  for (int r = 0; r < 8; ++r) {
    store_c(C, addend, bias, ldc, r0 + r,      c0,      N, act, acc00[r]);
    store_c(C, addend, bias, ldc, r0 + r,      c0 + 16, N, act, acc01[r]);
    store_c(C, addend, bias, ldc, r0 + r + 16, c0,      N, act, acc10[r]);
    store_c(C, addend, bias, ldc, r0 + r + 16, c0 + 16, N, act, acc11[r]);
  }
}

// ---------------------------------------------------------------------------
// LayerNorm over last dim (512). One 256-thread block per token row.
// ---------------------------------------------------------------------------
__global__ void ln_kernel(const float* __restrict__ x, const float* __restrict__ g,
                          const float* __restrict__ b, float* __restrict__ o) {
  __shared__ float red[256];
  const int t = blockIdx.x, tid = threadIdx.x;
  const float* xr = x + (size_t)t * EMBED;
  float v0 = xr[tid], v1 = xr[tid + 256];
  red[tid] = v0 + v1;
  __syncthreads();
  for (int s = 128; s; s >>= 1) { if (tid < s) red[tid] += red[tid + s]; __syncthreads(); }
  float mean = red[0] * (1.f / EMBED);
  __syncthreads();
  float d0 = v0 - mean, d1 = v1 - mean;
  red[tid] = d0 * d0 + d1 * d1;
  __syncthreads();
  for (int s = 128; s; s >>= 1) { if (tid < s) red[tid] += red[tid + s]; __syncthreads(); }
  float inv = rsqrtf(red[0] * (1.f / EMBED) + 1e-5f);
  o[(size_t)t * EMBED + tid]       = d0 * inv * g[tid]       + b[tid];
  o[(size_t)t * EMBED + tid + 256] = d1 * inv * g[tid + 256] + b[tid + 256];
}

// ---------------------------------------------------------------------------
// Gate: softmax over 5 expert logits, top-2, renormalized. One wave per token.
// ---------------------------------------------------------------------------
__global__ void gate_kernel(const float* __restrict__ xn, const float* __restrict__ gw,
                            float* __restrict__ w2, int* __restrict__ idx2) {
  const int t = blockIdx.x, lane = threadIdx.x;
  float s[NEXP] = {0.f, 0.f, 0.f, 0.f, 0.f};
  for (int i = lane; i < EMBED; i += 32) {
    float v = xn[(size_t)t * EMBED + i];
#pragma unroll
    for (int e = 0; e < NEXP; ++e) s[e] += v * gw[e * EMBED + i];
  }
#pragma unroll
  for (int e = 0; e < NEXP; ++e)
    for (int o = 16; o; o >>= 1) s[e] += __shfl_xor(s[e], o, 32);
  if (lane == 0) {
    float mx = s[0];
#pragma unroll
    for (int e = 1; e < NEXP; ++e) mx = fmaxf(mx, s[e]);
    float p[NEXP], sum = 0.f;
#pragma unroll
    for (int e = 0; e < NEXP; ++e) { p[e] = __expf(s[e] - mx); sum += p[e]; }
#pragma unroll
    for (int e = 0; e < NEXP; ++e) p[e] /= sum;
    int i0 = 0;
#pragma unroll
    for (int e = 1; e < NEXP; ++e) if (p[e] > p[i0]) i0 = e;
    int i1 = (i0 == 0) ? 1 : 0;
#pragma unroll
    for (int e = 0; e < NEXP; ++e) if (e != i0 && p[e] > p[i1]) i1 = e;
    float denom = fmaxf(p[i0] + p[i1], 1e-9f);
    w2[t * 2] = p[i0] / denom;  w2[t * 2 + 1] = p[i1] / denom;
    idx2[t * 2] = i0;           idx2[t * 2 + 1] = i1;
  }
}

// ---------------------------------------------------------------------------
// Causal depthwise conv1d (DCONV=4) + bias + SiLU. xi = xz[:, 0:DIN].
// ---------------------------------------------------------------------------
__global__ void conv_silu_kernel(const float* __restrict__ xz,
                                 const float* __restrict__ cw,
                                 const float* __restrict__ cb,
                                 float* __restrict__ u) {
  const int idx = blockIdx.x * blockDim.x + threadIdx.x;   // over NTOK*DIN
  const int d  = idx & (DIN - 1);
  const int bl = idx >> 10;
  const int l  = bl & (LL - 1);
  const int b  = bl >> 8;
  float acc = cb[d];
#pragma unroll
  for (int j = 0; j < DCONV; ++j) {
    int ls = l - (DCONV - 1) + j;
    if (ls >= 0) acc += xz[((size_t)(b * LL + ls)) * (2 * DIN) + d] * cw[d * DCONV + j];
  }
  u[(size_t)bl * DIN + d] = silu_f(acc);
}

// ---------------------------------------------------------------------------
// Selective scan: one wave per (b,d) channel; 32 lanes x 2 states = DSTATE.
// Fuses the + u*D skip and * silu(z) epilogue at write time.
// ---------------------------------------------------------------------------
__global__ void scan_kernel(const float* __restrict__ delta,  // (NTOK, DIN)
                            const float* __restrict__ u,      // (NTOK, DIN)
                            const float* __restrict__ dbc,    // (NTOK, 160): [dt|B|C]
                            const float* __restrict__ xz,     // (NTOK, 2*DIN): z at +DIN
                            const float* __restrict__ A_log,  // (DIN, DSTATE)
                            const float* __restrict__ Dsk,    // (DIN)
                            float* __restrict__ y)            // (NTOK, DIN)
{
  const int wid  = (blockIdx.x * blockDim.x + threadIdx.x) >> 5;   // 0..BB*DIN-1
  const int lane = threadIdx.x & 31;
  const int b = wid >> 10;
  const int d = wid & (DIN - 1);
  const int s0 = lane, s1 = lane + 32;

  const float A0 = -__expf(A_log[d * DSTATE + s0]);
  const float A1 = -__expf(A_log[d * DSTATE + s1]);
  const float Dv = Dsk[d];
  float h0 = 0.f, h1 = 0.f;

  for (int l = 0; l < LL; ++l) {
    const size_t bl = (size_t)(b * LL + l);
    float dt = delta[bl * DIN + d];
    float uu = u[bl * DIN + d];
    const float* row = dbc + bl * (DTRANK + 2 * DSTATE);
    float b0 = row[DTRANK + s0],          b1 = row[DTRANK + s1];
    float c0 = row[DTRANK + DSTATE + s0], c1 = row[DTRANK + DSTATE + s1];
    float du = dt * uu;
    h0 = h0 * __expf(dt * A0) + du * b0;
    h1 = h1 * __expf(dt * A1) + du * b1;
    float yp = h0 * c0 + h1 * c1;
    for (int o = 16; o; o >>= 1) yp += __shfl_xor(yp, o, 32);
    if (lane == 0) {
      float z = xz[bl * (2 * DIN) + DIN + d];
      y[bl * DIN + d] = (yp + uu * Dv) * silu_f(z);
    }
  }
}

// ---------------------------------------------------------------------------
// Top-2 combine + residual: x1 = x + w0*eo[i0] + w1*eo[i1]
// ---------------------------------------------------------------------------
__global__ void combine_kernel(const float* __restrict__ x, const float* __restrict__ eo,
                               const float* __restrict__ w2, const int* __restrict__ idx2,
                               float* __restrict__ x1) {
  const int i = blockIdx.x * blockDim.x + threadIdx.x;   // over NTOK*EMBED
  const int t = i >> 9;
  const size_t estride = (size_t)NTOK * EMBED;
  float f = w2[t * 2]     * eo[(size_t)idx2[t * 2]     * estride + i]
          + w2[t * 2 + 1] * eo[(size_t)idx2[t * 2 + 1] * estride + i];
  x1[i] = x[i] + f;
}

// ---------------------------------------------------------------------------
extern "C" void kernel_launch(void* const* d_in, const int* in_sizes, int n_in,
                              void* d_out, int out_size, void* d_ws, size_t ws_size,
                              hipStream_t stream) {
  (void)in_sizes; (void)n_in; (void)out_size; (void)ws_size;
  const float* x      = (const float*)d_in[0];
  const float* gate_w = (const float*)d_in[1];
  const float* ln1_g  = (const float*)d_in[2];
  const float* ln1_b  = (const float*)d_in[3];
  const float* ln2_g  = (const float*)d_in[4];
  const float* ln2_b  = (const float*)d_in[5];
  const float* ffn_w1 = (const float*)d_in[6];
  const float* ffn_b1 = (const float*)d_in[7];
  const float* ffn_w2 = (const float*)d_in[8];
  const float* ffn_b2 = (const float*)d_in[9];
  const float* in_w   = (const float*)d_in[10];
  const float* conv_w = (const float*)d_in[11];
  const float* conv_b = (const float*)d_in[12];
  const float* xp_w   = (const float*)d_in[13];
  const float* dt_w   = (const float*)d_in[14];
  const float* dt_b   = (const float*)d_in[15];
  const float* A_log  = (const float*)d_in[16];
  const float* D_skip = (const float*)d_in[17];
  const float* out_w  = (const float*)d_in[18];
  float* out = (float*)d_out;

  // Workspace layout (floats)
  float* ws = (float*)d_ws;
  size_t off = 0;
  float* xn    = ws + off; off += (size_t)NTOK * EMBED;
  float* w2    = ws + off; off += (size_t)NTOK * 2;
  int*   idx2  = (int*)(ws + off); off += (size_t)NTOK * 2;
  float* xzb   = ws + off; off += (size_t)NEXP * NTOK * 2 * DIN;
  float* ub    = ws + off; off += (size_t)NEXP * NTOK * DIN;
  float* dbcb  = ws + off; off += (size_t)NEXP * NTOK * (DTRANK + 2 * DSTATE);
  float* delb  = ws + off; off += (size_t)NEXP * NTOK * DIN;
  float* yb    = ws + off; off += (size_t)NEXP * NTOK * DIN;
  float* eob   = ws + off; off += (size_t)NEXP * NTOK * EMBED;
  float* x1    = ws + off; off += (size_t)NTOK * EMBED;
  float* hb    = ws + off; off += (size_t)NTOK * EMBED;
  float* gb    = ws + off; off += (size_t)NTOK * 2 * EMBED;

  ln_kernel<<<NTOK, 256, 0, stream>>>(x, ln1_g, ln1_b, xn);
  gate_kernel<<<NTOK, 32, 0, stream>>>(xn, gate_w, w2, idx2);

  for (int e = 0; e < NEXP; ++e) {
    float* xz_e  = xzb  + (size_t)e * NTOK * 2 * DIN;
    float* u_e   = ub   + (size_t)e * NTOK * DIN;
    float* dbc_e = dbcb + (size_t)e * NTOK * (DTRANK + 2 * DSTATE);
    float* del_e = delb + (size_t)e * NTOK * DIN;
    float* y_e   = yb   + (size_t)e * NTOK * DIN;
    float* eo_e  = eob  + (size_t)e * NTOK * EMBED;

    // in_proj: xz = xn @ in_w[e]^T   (512x2048, K=512)
    wmma_gemm_kernel<<<dim3(2 * DIN / 64, NTOK / 64), 128, 0, stream>>>(
        xn, EMBED, in_w + (size_t)e * 2 * DIN * EMBED, EMBED,
        nullptr, nullptr, xz_e, 2 * DIN, 2 * DIN, EMBED, 0);

    // depthwise causal conv + bias + silu -> u
    conv_silu_kernel<<<(NTOK * DIN) / 256, 256, 0, stream>>>(
        xz_e, conv_w + (size_t)e * DIN * DCONV, conv_b + (size_t)e * DIN, u_e);

    // xp: dbc = u @ xp_w[e]^T   (512x160, K=1024; N=160 predicated)
    wmma_gemm_kernel<<<dim3(3, NTOK / 64), 128, 0, stream>>>(
        u_e, DIN, xp_w + (size_t)e * (DTRANK + 2 * DSTATE) * DIN, DIN,
        nullptr, nullptr, dbc_e, DTRANK + 2 * DSTATE, DTRANK + 2 * DSTATE, DIN, 0);

    // delta = softplus(dt @ dt_w[e]^T + dt_b)  (512x1024, K=32; A strided in dbc)
    wmma_gemm_kernel<<<dim3(DIN / 64, NTOK / 64), 128, 0, stream>>>(
        dbc_e, DTRANK + 2 * DSTATE, dt_w + (size_t)e * DIN * DTRANK, DTRANK,
        dt_b + (size_t)e * DIN, nullptr, del_e, DIN, DIN, DTRANK, 1);

    // selective scan (+ u*D, * silu(z) fused): one wave per (b,d)
    scan_kernel<<<(BB * DIN * 32) / 256, 256, 0, stream>>>(
        del_e, u_e, dbc_e, xz_e,
        A_log + (size_t)e * DIN * DSTATE, D_skip + (size_t)e * DIN, y_e);

    // out_proj: eo = y @ out_w[e]^T   (512x512, K=1024)
    wmma_gemm_kernel<<<dim3(EMBED / 64, NTOK / 64), 128, 0, stream>>>(
        y_e, DIN, out_w + (size_t)e * EMBED * DIN, DIN,
        nullptr, nullptr, eo_e, EMBED, EMBED, DIN, 0);
  }

  combine_kernel<<<(NTOK * EMBED) / 256, 256, 0, stream>>>(x, eob, w2, idx2, x1);
  ln_kernel<<<NTOK, 256, 0, stream>>>(x1, ln2_g, ln2_b, hb);
  // FFN1 + exact GELU   (512x1024, K=512)
  wmma_gemm_kernel<<<dim3(2 * EMBED / 64, NTOK / 64), 128, 0, stream>>>(
      hb, EMBED, ffn_w1, EMBED, ffn_b1, nullptr, gb, 2 * EMBED, 2 * EMBED, EMBED, 2);
  // FFN2 + bias + residual -> out   (512x512, K=1024)
  wmma_gemm_kernel<<<dim3(EMBED / 64, NTOK / 64), 128, 0, stream>>>(
      gb, 2 * EMBED, ffn_w2, 2 * EMBED, ffn_b2, x1, out, EMBED, EMBED, 2 * EMBED, 0);
}